// TransformerAttentionSepModule_59390807769626
// MI455X (gfx1250) — compile-verified
//
#include <hip/hip_runtime.h>
#include <hip/hip_bf16.h>

// ---------------------------------------------------------------------------
// Graph transformer attention (GAT-style) for MI455X / gfx1250, wave32.
//   qkv   = x @ W_qkv + b_qkv                    (f32 WMMA 16x16x4)
//   s[e,h]= <k[src],q[dst]> * COEF               (per-edge wave, shfl reduce)
//   smax  = segment_max(s, dst)                  (ordered-uint atomicMax)
//   ex    = exp(s - smax[dst]); denom = segsum   (f32 atomicAdd)
//   msg   = segsum(v[src] * ex/denom)            (f32 atomicAdd scatter)
//   out   = [x, msg] @ W_out + b_out             (f32 WMMA 16x16x4)
// ---------------------------------------------------------------------------

typedef __attribute__((ext_vector_type(2))) float v2f;
typedef __attribute__((ext_vector_type(8))) float v8f;

#define N_NODES   100000
#define N_EDGES   3200000
#define DMODEL    256
#define NHEAD     8
#define DHEAD     32
#define QKV_COLS  768            // 3 * DMODEL
#define ATTN_COEF 0.17677669529663687f   // 1/sqrt(32)

// Order-preserving float <-> uint encoding for atomic max on floats.
__device__ __forceinline__ unsigned enc_f(float f) {
  unsigned u = __float_as_uint(f);
  return (u & 0x80000000u) ? ~u : (u | 0x80000000u);
}
__device__ __forceinline__ float dec_f(unsigned u) {
  return __uint_as_float((u & 0x80000000u) ? (u ^ 0x80000000u) : ~u);
}

// ---------------------------------------------------------------------------
// Kernel 0: initialize smax (encoded -inf), denom, message accumulators.
// ---------------------------------------------------------------------------
__global__ __launch_bounds__(256)
void init_kernel(unsigned* __restrict__ smax_enc, float* __restrict__ denom,
                 float* __restrict__ msg) {
  int i = blockIdx.x * 256 + threadIdx.x;
  if (i < N_NODES * NHEAD) {
    smax_enc[i] = 0x007FFFFFu;   // enc(-inf)
    denom[i]    = 0.0f;
  }
  if (i < N_NODES * DMODEL) msg[i] = 0.0f;
}

// ---------------------------------------------------------------------------
// Kernel 1: qkv = x @ W_qkv + b_qkv.
// One wave per 16x64 tile: 4 WMMA accumulators, K=256 in steps of 4.
// A 16x4 frag: lanes 0-15 -> K = {k0,k0+1}; lanes 16-31 -> K = {k0+2,k0+3}.
// B 4x16 frag: VGPR0 holds rows {k0, k0+2}, VGPR1 rows {k0+1, k0+3}.
// C 16x16:     VGPR i holds rows i (lanes 0-15) and i+8 (lanes 16-31).
// ---------------------------------------------------------------------------
__global__ __launch_bounds__(32)
void qkv_gemm_kernel(const float* __restrict__ x, const float* __restrict__ Wq,
                     const float* __restrict__ bq, float* __restrict__ qkv) {
  const int lane = threadIdx.x;
  const int l    = lane & 15;
  const int half = lane >> 4;
  const int row0 = blockIdx.x * 16;
  const int n0   = blockIdx.y * 64;

  v8f acc[4] = {};
  const float* arow = x + (size_t)(row0 + l) * DMODEL;

  for (int k0 = 0; k0 < DMODEL; k0 += 4) {
    const int kk = k0 + 2 * half;
    v2f a;
    a.x = arow[kk];
    a.y = arow[kk + 1];
#pragma unroll
    for (int j = 0; j < 4; ++j) {
      const int col = n0 + j * 16 + l;
      v2f b;
      b.x = Wq[(size_t)kk * QKV_COLS + col];
      b.y = Wq[(size_t)(kk + 1) * QKV_COLS + col];
      acc[j] = __builtin_amdgcn_wmma_f32_16x16x4_f32(
          false, a, false, b, (short)0, acc[j], false, false);
    }
  }

#pragma unroll
  for (int j = 0; j < 4; ++j) {
    const int col = n0 + j * 16 + l;
    const float bias = bq[col];
#pragma unroll
    for (int i = 0; i < 8; ++i) {
      const int r = row0 + i + 8 * half;
      qkv[(size_t)r * QKV_COLS + col] = acc[j][i] + bias;
    }
  }
}

// ---------------------------------------------------------------------------
// Kernel 2: edge scores + segment max.
// One wave per edge; lane = h*4 + i, each lane sums 8 elements of the
// 32-dim head dot product, reduced across the 4 lanes of the head.
// qkv row layout per head h: q at h*96, k at h*96+32, v at h*96+64.
// ---------------------------------------------------------------------------
__global__ __launch_bounds__(256)
void scores_kernel(const float* __restrict__ qkv, const int* __restrict__ src,
                   const int* __restrict__ dst, float* __restrict__ scores,
                   unsigned* __restrict__ smax_enc) {
  const int gid = blockIdx.x * 256 + threadIdx.x;
  const int e   = gid >> 5;
  if (e >= N_EDGES) return;           // wave-uniform (blockDim multiple of 32)
  const int lane = threadIdx.x & 31;
  const int h    = lane >> 2;
  const int i    = lane & 3;
  const int s = src[e];
  const int t = dst[e];

  const float* krow = qkv + (size_t)s * QKV_COLS + h * 96 + 32 + i * 8;
  const float* qrow = qkv + (size_t)t * QKV_COLS + h * 96 + i * 8;
  float sum = 0.0f;
#pragma unroll
  for (int j = 0; j < 8; ++j) sum += krow[j] * qrow[j];

  sum += __shfl_xor(sum, 1, 32);
  sum += __shfl_xor(sum, 2, 32);

  if (i == 0) {
    const float sc = sum * ATTN_COEF;
    scores[(size_t)e * NHEAD + h] = sc;
    atomicMax(smax_enc + (size_t)t * NHEAD + h, enc_f(sc));
  }
}

// ---------------------------------------------------------------------------
// Kernel 3: ex = exp(s - smax[dst]) (in place), denom = segment_sum(ex).
// One thread per (edge, head).
// ---------------------------------------------------------------------------
__global__ __launch_bounds__(256)
void exp_denom_kernel(const int* __restrict__ dst,
                      const unsigned* __restrict__ smax_enc,
                      float* __restrict__ ex, float* __restrict__ denom) {
  const int idx = blockIdx.x * 256 + threadIdx.x;
  if (idx >= N_EDGES * NHEAD) return;
  const int e = idx >> 3;
  const int h = idx & 7;
  const int t = dst[e];
  float m = dec_f(smax_enc[(size_t)t * NHEAD + h]);
  if (m < -3.0e38f) m = 0.0f;          // only hit for non-finite segment max
  const float v = __expf(ex[idx] - m);
  ex[idx] = v;
  atomicAdd(denom + (size_t)t * NHEAD + h, v);
}

// ---------------------------------------------------------------------------
// Kernel 4: message scatter.  One wave per edge; lane d = 0..31, loop heads.
// msg[dst, h*32 + d] += v[src, h, d] * ex[e,h] / denom[dst,h]
// ---------------------------------------------------------------------------
__global__ __launch_bounds__(256)
void message_kernel(const float* __restrict__ qkv, const int* __restrict__ src,
                    const int* __restrict__ dst, const float* __restrict__ ex,
                    const float* __restrict__ denom, float* __restrict__ msg) {
  const int gid = blockIdx.x * 256 + threadIdx.x;
  const int e   = gid >> 5;
  if (e >= N_EDGES) return;
  const int d = threadIdx.x & 31;
  const int s = src[e];
  const int t = dst[e];

  const float* vrow = qkv + (size_t)s * QKV_COLS + 64 + d;  // + h*96
  float*       mrow = msg + (size_t)t * DMODEL + d;         // + h*32
#pragma unroll
  for (int h = 0; h < NHEAD; ++h) {
    const float w = ex[(size_t)e * NHEAD + h] /
                    fmaxf(denom[(size_t)t * NHEAD + h], 1e-20f);
    atomicAdd(mrow + h * DHEAD, vrow[h * 96] * w);
  }
}

// ---------------------------------------------------------------------------
// Kernel 5: out = concat([x, msg]) @ W_out + b_out.  Same WMMA tiling, K=512.
// ---------------------------------------------------------------------------
__global__ __launch_bounds__(32)
void out_gemm_kernel(const float* __restrict__ x, const float* __restrict__ msg,
                     const float* __restrict__ Wo, const float* __restrict__ bo,
                     float* __restrict__ out) {
  const int lane = threadIdx.x;
  const int l    = lane & 15;
  const int half = lane >> 4;
  const int row0 = blockIdx.x * 16;
  const int n0   = blockIdx.y * 64;

  v8f acc[4] = {};
  const float* xrow = x + (size_t)(row0 + l) * DMODEL;
  const float* mrow = msg + (size_t)(row0 + l) * DMODEL;

  for (int k0 = 0; k0 < 2 * DMODEL; k0 += 4) {
    const int kk = k0 + 2 * half;
    v2f a;
    if (kk < DMODEL) {
      a.x = xrow[kk];
      a.y = xrow[kk + 1];
    } else {
      a.x = mrow[kk - DMODEL];
      a.y = mrow[kk - DMODEL + 1];
    }
#pragma unroll
    for (int j = 0; j < 4; ++j) {
      const int col = n0 + j * 16 + l;
      v2f b;
      b.x = Wo[(size_t)kk * DMODEL + col];
      b.y = Wo[(size_t)(kk + 1) * DMODEL + col];
      acc[j] = __builtin_amdgcn_wmma_f32_16x16x4_f32(
          false, a, false, b, (short)0, acc[j], false, false);
    }
  }

#pragma unroll
  for (int j = 0; j < 4; ++j) {
    const int col = n0 + j * 16 + l;
    const float bias = bo[col];
#pragma unroll
    for (int i = 0; i < 8; ++i) {
      const int r = row0 + i + 8 * half;
      out[(size_t)r * DMODEL + col] = acc[j][i] + bias;
    }
  }
}

// ---------------------------------------------------------------------------
// Host launcher.
// Inputs: 0=x, 1=W_qkv, 2=b_qkv, 3=W_out, 4=b_out, 5=src, 6=dst.
// ---------------------------------------------------------------------------
extern "C" void kernel_launch(void* const* d_in, const int* in_sizes, int n_in,
                              void* d_out, int out_size, void* d_ws, size_t ws_size,
                              hipStream_t stream) {
  const float* x   = (const float*)d_in[0];
  const float* Wq  = (const float*)d_in[1];
  const float* bq  = (const float*)d_in[2];
  const float* Wo  = (const float*)d_in[3];
  const float* bo  = (const float*)d_in[4];
  const int*   src = (const int*)d_in[5];
  const int*   dst = (const int*)d_in[6];
  float* out = (float*)d_out;

  // Workspace layout (floats / u32):
  float*    qkv   = (float*)d_ws;                                // N * 768
  float*    ex    = qkv + (size_t)N_NODES * QKV_COLS;            // E * 8
  unsigned* smax  = (unsigned*)(ex + (size_t)N_EDGES * NHEAD);   // N * 8
  float*    denom = (float*)(smax + (size_t)N_NODES * NHEAD);    // N * 8
  float*    msg   = denom + (size_t)N_NODES * NHEAD;             // N * 256

  // 0: init accumulators
  {
    int total = N_NODES * DMODEL;
    init_kernel<<<(total + 255) / 256, 256, 0, stream>>>(smax, denom, msg);
  }
  // 1: qkv projection (16x64 tile per wave; grid 6250 x 12)
  {
    dim3 grid(N_NODES / 16, QKV_COLS / 64);
    qkv_gemm_kernel<<<grid, 32, 0, stream>>>(x, Wq, bq, qkv);
  }
  // 2: edge scores + segment max (one wave per edge)
  scores_kernel<<<(N_EDGES * 32) / 256, 256, 0, stream>>>(qkv, src, dst, ex, smax);
  // 3: exp + segment sum of denominators
  exp_denom_kernel<<<(N_EDGES * NHEAD) / 256, 256, 0, stream>>>(dst, smax, ex, denom);
  // 4: weighted message scatter
  message_kernel<<<(N_EDGES * 32) / 256, 256, 0, stream>>>(qkv, src, dst, ex, denom, msg);
  // 5: output projection (grid 6250 x 4)
  {
    dim3 grid(N_NODES / 16, DMODEL / 64);
    out_gemm_kernel<<<grid, 32, 0, stream>>>(x, msg, Wo, bo, out);
  }
}